// SSIM_52338471469114
// MI455X (gfx1250) — compile-verified
//
#include <hip/hip_runtime.h>

// CDNA5 / gfx1250: wave32, WMMA f32 16x16x4 (fp32 matrix pipe).
typedef __attribute__((ext_vector_type(2))) float v2f;
typedef __attribute__((ext_vector_type(8))) float v8f;

#define WMMA4(A, B, C) \
  __builtin_amdgcn_wmma_f32_16x16x4_f32(false, (A), false, (B), (short)0, (C), false, false)

// SSIM constants (DATA_RANGE = 1.0)
#define SSIM_C1 0.0001f
#define SSIM_C2 0.0009f

// Problem: N=16, C=3, H=W=512 fp32.
// Per-wave tile: 16 wide x 32 tall.  Per plane: 32 col-tiles x 16 row-tiles = 512.
#define IMG_H 512
#define IMG_W 512
#define N_PLANES 48
#define TILES_PER_PLANE 512
#define WAVES_PER_BLOCK 4
#define N_TILES (N_PLANES * TILES_PER_PLANE)     // 24576
#define N_BLOCKS (N_TILES / WAVES_PER_BLOCK)     // 6144
#define MEAN_SCALE (1.0f / (16.0f * 3.0f * 512.0f * 512.0f))

// Normalized 1D Gaussian (window=11, sigma=1.5), zero-padded so that
// GWP[15 + d] = g[d] for d in [0,10] and 0 elsewhere (d in [-15, 27]).
__constant__ float GWP[43] = {
  0.f, 0.f, 0.f, 0.f, 0.f, 0.f, 0.f, 0.f, 0.f, 0.f,
  0.f, 0.f, 0.f, 0.f, 0.f,
  0.00102837f, 0.00759878f, 0.03600079f, 0.10936073f, 0.21300540f,
  0.26601183f,
  0.21300540f, 0.10936073f, 0.03600079f, 0.00759878f, 0.00102837f,
  0.f, 0.f, 0.f, 0.f, 0.f, 0.f, 0.f, 0.f, 0.f, 0.f,
  0.f, 0.f, 0.f, 0.f, 0.f, 0.f, 0.f
};

__global__ __launch_bounds__(128) void ssim_wmma_kernel(
    const float* __restrict__ pred,
    const float* __restrict__ targ,
    float* __restrict__ out)
{
  // Per-wave staging of the 5 horizontally-blurred quantities:
  // 48 rows (32 outputs + -5..+5 vertical halo, padded to 3 row-tiles) x 16 cols.
  __shared__ float hb[WAVES_PER_BLOCK][5][48 * 16];
  __shared__ float bsum[WAVES_PER_BLOCK];

  const int wave = threadIdx.x >> 5;
  const int lane = threadIdx.x & 31;
  const int half = lane >> 4;     // K-striping half (lanes 16-31 hold K+2/K+3)
  const int l16  = lane & 15;     // M (A rows) / N (B cols)

  // Banded weight fragments: value(j) = g[j - l16] (zero outside the band).
  // Identical per-lane values serve as the B matrix of the horizontal pass and
  // the A matrix of the vertical pass (same banded g[j-n], same K striping).
  v2f wf[7];
  #pragma unroll
  for (int kk = 0; kk < 7; ++kk) {
    const int j0 = 4 * kk + 2 * half;
    wf[kk][0] = GWP[j0 - l16 + 15];
    wf[kk][1] = GWP[j0 - l16 + 16];
  }

  // ---- tile assignment: one 16x32 output tile per wave ----
  const int tile  = blockIdx.x * WAVES_PER_BLOCK + wave;
  const int plane = tile / TILES_PER_PLANE;
  const int tip   = tile % TILES_PER_PLANE;
  const int y0 = (tip >> 5) << 5;        // row-tile * 32
  const int x0 = (tip & 31) << 4;        // col-tile * 16
  const float* __restrict__ P = pred + (size_t)plane * (IMG_H * IMG_W);
  const float* __restrict__ T = targ + (size_t)plane * (IMG_H * IMG_W);

  // ================= horizontal pass (row blur) via WMMA =================
  // out16x16 = A(16 rows x 28 in-cols) * Wband(28x16); three row-tiles cover
  // rows y0-5 .. y0+42 (incl. vertical halo).  Branch-free zero padding via
  // clamped indices + 0/1 masks.
  #pragma unroll
  for (int rt = 0; rt < 3; ++rt) {
    v8f accP = {}, accT = {}, accPP = {}, accTT = {}, accPT = {};
    const int y  = y0 - 5 + rt * 16 + l16;
    const int yc = min(max(y, 0), IMG_H - 1);
    const float ym = (y == yc) ? 1.f : 0.f;
    const float* prow = P + (ptrdiff_t)yc * IMG_W;
    const float* trow = T + (ptrdiff_t)yc * IMG_W;
    #pragma unroll
    for (int kk = 0; kk < 7; ++kk) {
      const int x   = x0 - 5 + 4 * kk + 2 * half;  // A layout: v0=K(+0/+2), v1=K(+1/+3)
      const int xc0 = min(max(x, 0), IMG_W - 1);
      const int xc1 = min(max(x + 1, 0), IMG_W - 1);
      const float m0 = (x == xc0) ? ym : 0.f;
      const float m1 = (x + 1 == xc1) ? ym : 0.f;
      const float p0 = prow[xc0] * m0, p1 = prow[xc1] * m1;
      const float t0 = trow[xc0] * m0, t1 = trow[xc1] * m1;
      const v2f aP  = {p0,      p1};
      const v2f aT  = {t0,      t1};
      const v2f aPP = {p0 * p0, p1 * p1};
      const v2f aTT = {t0 * t0, t1 * t1};
      const v2f aPT = {p0 * t0, p1 * t1};
      accP  = WMMA4(aP,  wf[kk], accP);
      accT  = WMMA4(aT,  wf[kk], accT);
      accPP = WMMA4(aPP, wf[kk], accPP);
      accTT = WMMA4(aTT, wf[kk], accTT);
      accPT = WMMA4(aPT, wf[kk], accPT);
    }
    // D layout: VGPR r -> row (r + 8*half), col l16. Stage into LDS.
    #pragma unroll
    for (int r = 0; r < 8; ++r) {
      const int idx = (rt * 16 + r + 8 * half) * 16 + l16;
      hb[wave][0][idx] = accP[r];
      hb[wave][1][idx] = accT[r];
      hb[wave][2][idx] = accPP[r];
      hb[wave][3][idx] = accTT[r];
      hb[wave][4][idx] = accPT[r];
    }
  }
  // Wave-private LDS region; DS ops from the same wave stay in order.

  // ================= vertical pass (column blur) + SSIM =================
  float s = 0.f;
  #pragma unroll
  for (int mt = 0; mt < 2; ++mt) {
    v8f bP = {}, bT = {}, bPP = {}, bTT = {}, bPT = {};
    #pragma unroll
    for (int kk = 0; kk < 7; ++kk) {
      // B layout: v0 = hb row (16*mt + 4kk + 0/+2), v1 = next row, col l16.
      const int i0 = (mt * 16 + 4 * kk + 2 * half) * 16 + l16;
      const v2f dP  = { hb[wave][0][i0], hb[wave][0][i0 + 16] };
      const v2f dT  = { hb[wave][1][i0], hb[wave][1][i0 + 16] };
      const v2f dPP = { hb[wave][2][i0], hb[wave][2][i0 + 16] };
      const v2f dTT = { hb[wave][3][i0], hb[wave][3][i0 + 16] };
      const v2f dPT = { hb[wave][4][i0], hb[wave][4][i0 + 16] };
      bP  = WMMA4(wf[kk], dP,  bP);
      bT  = WMMA4(wf[kk], dT,  bT);
      bPP = WMMA4(wf[kk], dPP, bPP);
      bTT = WMMA4(wf[kk], dTT, bTT);
      bPT = WMMA4(wf[kk], dPT, bPT);
    }
    #pragma unroll
    for (int r = 0; r < 8; ++r) {
      const float mu1 = bP[r], mu2 = bT[r];
      const float mu1s = mu1 * mu1, mu2s = mu2 * mu2, mu12 = mu1 * mu2;
      const float s1  = bPP[r] - mu1s;
      const float s2  = bTT[r] - mu2s;
      const float s12 = bPT[r] - mu12;
      const float num = (2.f * mu12 + SSIM_C1) * (2.f * s12 + SSIM_C2);
      const float den = (mu1s + mu2s + SSIM_C1) * (s1 + s2 + SSIM_C2);
      s += num / den;
    }
  }

  // ================= reduction =================
  #pragma unroll
  for (int off = 16; off > 0; off >>= 1)
    s += __shfl_down(s, off, 32);
  if (lane == 0) bsum[wave] = s;
  __syncthreads();
  if (threadIdx.x == 0) {
    const float t = bsum[0] + bsum[1] + bsum[2] + bsum[3];
    atomicAdd(out, t * MEAN_SCALE);
  }
}

extern "C" void kernel_launch(void* const* d_in, const int* in_sizes, int n_in,
                              void* d_out, int out_size, void* d_ws, size_t ws_size,
                              hipStream_t stream) {
  (void)in_sizes; (void)n_in; (void)out_size; (void)d_ws; (void)ws_size;
  const float* pred = (const float*)d_in[0];
  const float* targ = (const float*)d_in[1];
  float* out = (float*)d_out;

  hipMemsetAsync(out, 0, sizeof(float), stream);   // graph-capturable
  ssim_wmma_kernel<<<dim3(N_BLOCKS), dim3(128), 0, stream>>>(pred, targ, out);
}